// QuantumFraudDetectionHybrid_65481071399896
// MI455X (gfx1250) — compile-verified
//
#include <hip/hip_runtime.h>

typedef float v2f __attribute__((ext_vector_type(2)));
typedef float v8f __attribute__((ext_vector_type(8)));

// gfx1250 hardware tanh (TRANS op) with safe fallback
#if __has_builtin(__builtin_amdgcn_tanhf)
#define FAST_TANH(x) __builtin_amdgcn_tanhf(x)
#elif __has_builtin(__builtin_amdgcn_tanh_f32)
#define FAST_TANH(x) __builtin_amdgcn_tanh_f32(x)
#else
#define FAST_TANH(x) tanhf(x)
#endif

static __device__ __forceinline__ v8f wmma4(v2f a, v2f b, v8f c) {
  // (neg_a, A, neg_b, B, c_mod, C, reuse_a, reuse_b) -> v_wmma_f32_16x16x4_f32
  return __builtin_amdgcn_wmma_f32_16x16x4_f32(false, a, false, b, (short)0, c,
                                               false, false);
}

static __device__ __forceinline__ v8f cmat(float e0, float e1) {
  v8f r;
  r[0] = e0; r[1] = e1;
  r[2] = 0.f; r[3] = 0.f; r[4] = 0.f; r[5] = 0.f; r[6] = 0.f; r[7] = 0.f;
  return r;
}

__global__ __launch_bounds__(256) void qfd_hybrid_kernel(
    const float* __restrict__ x,
    const float* __restrict__ W1, const float* __restrict__ b1,
    const float* __restrict__ s1, const float* __restrict__ sh1,
    const float* __restrict__ W2, const float* __restrict__ b2,
    const float* __restrict__ s2, const float* __restrict__ sh2,
    const float* __restrict__ vp, const float* __restrict__ hw,
    const float* __restrict__ hb, float* __restrict__ out, int n) {
  const int lane = (int)(threadIdx.x & 31u);  // wave32

  // ---- uniform parameter loads (broadcast, cached) ----
  const float w100 = W1[0], w101 = W1[1], w110 = W1[2], w111 = W1[3];
  const float b10 = b1[0], b11 = b1[1];
  const float s10 = s1[0], s11 = s1[1], q10 = sh1[0], q11 = sh1[1];
  const float w200 = W2[0], w201 = W2[1], w210 = W2[2], w211 = W2[3];
  const float b20 = b2[0], b21 = b2[1];
  const float s20 = s2[0], s21 = s2[1], q20 = sh2[0], q21 = sh2[1];
  const float ct0 = __cosf(vp[0]), ct1 = __cosf(vp[2]);
  const float hw0 = hw[0] * ct0, hw1 = hw[1] * ct1;  // fold cos(theta) in head
  const float hb0 = hb[0];

  // ---- A matrices: 16x4 f32, row m in lane m (VGPR0=K0, VGPR1=K1);
  //      lanes 16-31 hold the zero-padded K=2,3 columns. ----
  v2f a1, a2v, a3;
  a1[0]  = (lane == 0) ? w100 : ((lane == 1) ? w110 : 0.f);
  a1[1]  = (lane == 0) ? w101 : ((lane == 1) ? w111 : 0.f);
  a2v[0] = (lane == 0) ? w200 : ((lane == 1) ? w210 : 0.f);
  a2v[1] = (lane == 0) ? w201 : ((lane == 1) ? w211 : 0.f);
  a3[0]  = (lane == 0) ? hw0 : 0.f;
  a3[1]  = (lane == 0) ? hw1 : 0.f;

  // ---- C matrices: C/D layout puts row M=r in VGPR r, lanes 0-15 ----
  const bool lo = lane < 16;
  const v8f c1 = cmat(lo ? b10 : 0.f, lo ? b11 : 0.f);
  const v8f c2 = cmat(lo ? b20 : 0.f, lo ? b21 : 0.f);
  const v8f c3 = cmat(lo ? hb0 : 0.f, 0.f);

  const v2f* __restrict__ x2 = (const v2f*)x;
  const int ntiles = n >> 5;  // 32 samples per wave-iteration

  // Wave-uniform tile index via readfirstlane: loop condition becomes a
  // scalar branch, so EXEC is provably all-1s at every WMMA (ISA §7.12).
  const int wid =
      __builtin_amdgcn_readfirstlane((int)((blockIdx.x * blockDim.x + threadIdx.x) >> 5));
  const int nw = (int)((gridDim.x * blockDim.x) >> 5);

  for (int t = wid; t < ntiles; t += nw) {
    const int base = t << 5;
    // streamed once -> non-temporal; coalesced 256B per wave
    v2f xv = __builtin_nontemporal_load(x2 + (base + lane));

    // prefetch this wave's next tile (clamped; speculative prefetch is safe)
    long pn = (t + nw < ntiles) ? (((long)(t + nw) << 5) + lane)
                                : ((long)base + lane);
    __builtin_prefetch((const void*)(x2 + pn), 0, 0);

    // chain B = samples 16-31 moved into lanes 0-15
    v2f xw;
    xw[0] = __shfl_xor(xv[0], 16, 32);
    xw[1] = __shfl_xor(xv[1], 16, 32);

    // Layer 1: D = W1 * X^T + b1  (lanes 16-31 of B are K=2/3 rows; the
    // matching A columns are zero so their finite garbage contributes 0)
    v8f dA = wmma4(a1, xv, c1);
    v8f dB = wmma4(a1, xw, c1);

    v2f hA, hB;  // tanh(.)*s1 + sh1 per feature-VGPR — feeds next B directly
    hA[0] = fmaf(FAST_TANH(dA[0]), s10, q10);
    hA[1] = fmaf(FAST_TANH(dA[1]), s11, q11);
    hB[0] = fmaf(FAST_TANH(dB[0]), s10, q10);
    hB[1] = fmaf(FAST_TANH(dB[1]), s11, q11);

    // Layer 2
    dA = wmma4(a2v, hA, c2);
    dB = wmma4(a2v, hB, c2);

    v2f gA, gB;  // cos(tanh(.)*s2 + sh2); cos(theta) already folded into a3
    gA[0] = __cosf(fmaf(FAST_TANH(dA[0]), s20, q20));
    gA[1] = __cosf(fmaf(FAST_TANH(dA[1]), s21, q21));
    gB[0] = __cosf(fmaf(FAST_TANH(dB[0]), s20, q20));
    gB[1] = __cosf(fmaf(FAST_TANH(dB[1]), s21, q21));

    // Head: D row 0 (lanes 0-15) = outputs of 16 samples
    dA = wmma4(a3, gA, c3);
    dB = wmma4(a3, gB, c3);

    // pack both 16-sample chains into one full-wave 128B NT store
    const float oBs = __shfl_xor(dB[0], 16, 32);
    const float o = lo ? dA[0] : oBs;
    __builtin_nontemporal_store(o, out + (base + lane));
  }

  // ---- scalar tail for batch not a multiple of 32 (none for 2^23) ----
  const int done = ntiles << 5;
  const int tid = (int)(blockIdx.x * blockDim.x + threadIdx.x);
  if (tid < n - done) {
    const int s = done + tid;
    const float x0 = x[2 * s], x1 = x[2 * s + 1];
    const float h0 = fmaf(FAST_TANH(w100 * x0 + w101 * x1 + b10), s10, q10);
    const float h1 = fmaf(FAST_TANH(w110 * x0 + w111 * x1 + b11), s11, q11);
    const float g0 = __cosf(fmaf(FAST_TANH(w200 * h0 + w201 * h1 + b20), s20, q20));
    const float g1 = __cosf(fmaf(FAST_TANH(w210 * h0 + w211 * h1 + b21), s21, q21));
    out[s] = g0 * hw0 + g1 * hw1 + hb0;
  }
}

extern "C" void kernel_launch(void* const* d_in, const int* in_sizes, int n_in,
                              void* d_out, int out_size, void* d_ws,
                              size_t ws_size, hipStream_t stream) {
  (void)in_sizes; (void)n_in; (void)d_ws; (void)ws_size;
  const float* x   = (const float*)d_in[0];
  const float* W1  = (const float*)d_in[1];
  const float* b1  = (const float*)d_in[2];
  const float* s1  = (const float*)d_in[3];
  const float* sh1 = (const float*)d_in[4];
  const float* W2  = (const float*)d_in[5];
  const float* b2  = (const float*)d_in[6];
  const float* s2  = (const float*)d_in[7];
  const float* sh2 = (const float*)d_in[8];
  const float* vp  = (const float*)d_in[9];
  const float* hw  = (const float*)d_in[10];
  const float* hb  = (const float*)d_in[11];
  float* out = (float*)d_out;

  const int n = out_size;  // output is [B,1]
  if (n <= 0) return;
  // 3072 blocks x 256 threads = 24576 waves -> ~10.7 tiles/wave for 2^23
  // samples; amortizes per-wave A/C setup while saturating HBM.
  const int blocks = 3072;
  qfd_hybrid_kernel<<<blocks, 256, 0, stream>>>(
      x, W1, b1, s1, sh1, W2, b2, s2, sh2, vp, hw, hb, out, n);
}